// ClassicalRayMarcher_14998025797803
// MI455X (gfx1250) — compile-verified
//
#include <hip/hip_runtime.h>
#include <math.h>

typedef __attribute__((ext_vector_type(2))) float v2f;
typedef __attribute__((ext_vector_type(8))) float v8f;

#define INF_DELTA 1.0e10f
#define EPS_T     1e-10f

#define SAMPLES        64
#define RAYS_PER_BLOCK 64
#define RAYS_PER_WAVE  16
#define LDS_STRIDE     68   // 64 + 4 pad: conflict-free feeds (bank = (4n + s) & 63)

// accurate-enough log1p(t) for t in (0,1] using native ops only:
//  - t < 0.1 : 4-term alternating Taylor (abs err ~ t^5/5 <= 2e-6)
//  - t >= 0.1: native log, argument >= 1.1 so relative error is ~ulp-level
__device__ __forceinline__ float fast_log1p(float t) {
    float poly = t * (1.0f + t * (-0.5f + t * (0.33333333f - 0.25f * t)));
    float lg   = __logf(1.0f + t);
    return (t < 0.1f) ? poly : lg;
}

__global__ __launch_bounds__(128) void raymarch_wmma_kernel(
    const float* __restrict__ colors,     // [Nray, 64, 3]
    const float* __restrict__ densities,  // [Nray, 64]
    const float* __restrict__ depths,     // [Nray, 64]
    float* __restrict__ outRgb,           // [Nray, 3]
    float* __restrict__ outDepth,         // [Nray]
    float* __restrict__ outWeights,       // [Nray, 64]
    float* __restrict__ outFt)            // [Nray]
{
    __shared__ float ldsQ [RAYS_PER_BLOCK * LDS_STRIDE]; // q = 1 - alpha
    __shared__ float ldsLv[RAYS_PER_BLOCK * LDS_STRIDE]; // log(q + eps)

    const int tid  = threadIdx.x;
    const int wave = tid >> 5;
    const int lane = tid & 31;
    const int rayBlock0 = blockIdx.x * RAYS_PER_BLOCK;

    // ---------------- Stage: coalesced loads, compute q and log(q+eps) into LDS
    {
        const int rLocal = (wave * RAYS_PER_WAVE) + (lane >> 1); // block-local ray
        const int half   = lane & 1;                             // sample half (0..31 / 32..63)
        const long long gbase = (long long)(rayBlock0 + rLocal) * SAMPLES + half * 32;
        const float4* dp4 = (const float4*)(depths    + gbase);
        const float4* dn4 = (const float4*)(densities + gbase);
        float4 dv[8], nv[8];
        #pragma unroll
        for (int i = 0; i < 8; ++i) dv[i] = dp4[i];
        #pragma unroll
        for (int i = 0; i < 8; ++i) nv[i] = dn4[i];

        float dflat[33];
        #pragma unroll
        for (int i = 0; i < 8; ++i) {
            dflat[4*i+0] = dv[i].x; dflat[4*i+1] = dv[i].y;
            dflat[4*i+2] = dv[i].z; dflat[4*i+3] = dv[i].w;
        }
        // depths[ray][32] for the low-half lane comes from the partner lane's first element
        dflat[32] = __shfl(dv[0].x, lane | 1, 32);

        float nflat[32];
        #pragma unroll
        for (int i = 0; i < 8; ++i) {
            nflat[4*i+0] = nv[i].x; nflat[4*i+1] = nv[i].y;
            nflat[4*i+2] = nv[i].z; nflat[4*i+3] = nv[i].w;
        }

        #pragma unroll
        for (int i = 0; i < 32; ++i) {
            const int s = half * 32 + i;
            float delta = (s == SAMPLES - 1) ? INF_DELTA : (dflat[i+1] - dflat[i]);
            float x  = nflat[i];
            // softplus(x) = max(x,0) + log1p(exp(-|x|)), native exp + guarded log1p
            float t  = __expf(-fabsf(x));
            float sp = fmaxf(x, 0.0f) + fast_log1p(t);
            float q  = __expf(-delta * sp);                       // = 1 - alpha
            float lv = __logf(q + EPS_T);
            const int a = rLocal * LDS_STRIDE + s;
            ldsQ [a] = q;
            ldsLv[a] = lv;
        }
    }
    __syncthreads();

    // ---------------- WMMA scan: Y = L * X (+carry via C accumulator), 16 rays/wave
    const int  n   = lane & 15;           // ray column / A-matrix row (M)
    const bool hi  = lane >= 16;
    const int  off = hi ? 2 : 0;          // K-lane split per documented 16x4 f32 A layout
    const int  rLocal = wave * RAYS_PER_WAVE + n;
    const long long rayG = (long long)rayBlock0 + rLocal;

    // A = 16x16 lower-triangular ones, split into four 16x4 K-chunks
    v2f A[4];
    #pragma unroll
    for (int k = 0; k < 4; ++k) {
        const int base = 4 * k + off;
        A[k].x = (n >= base    ) ? 1.0f : 0.0f;
        A[k].y = (n >= base + 1) ? 1.0f : 0.0f;
    }

    float T = 0.0f;                        // running log-transmittance (per ray)
    float rgb0 = 0.f, rgb1 = 0.f, rgb2 = 0.f, dAcc = 0.f;

    #pragma unroll
    for (int b = 0; b < 4; ++b) {
        v8f c;
        #pragma unroll
        for (int j = 0; j < 8; ++j) c[j] = T;  // fold carry into accumulator

        #pragma unroll
        for (int k = 0; k < 4; ++k) {
            const int s0 = 16 * b + 4 * k + off;
            v2f Bv;
            Bv.x = ldsLv[rLocal * LDS_STRIDE + s0];
            Bv.y = ldsLv[rLocal * LDS_STRIDE + s0 + 1];
            c = __builtin_amdgcn_wmma_f32_16x16x4_f32(
                    false, A[k], false, Bv, (short)0, c, false, false);
        }
        // inclusive total of this block (row M=15 lives in VGPR7, lanes 16..31)
        const float blockTot = __shfl(c[7], 16 + n, 32);

        // D layout: lane owns samples sb..sb+7 of ray rayG
        const int sb = 16 * b + (hi ? 8 : 0);

        const float4* c4 = (const float4*)(colors + rayG * (SAMPLES * 3) + (long long)sb * 3);
        float cc[24];
        #pragma unroll
        for (int i = 0; i < 6; ++i) {
            float4 t = c4[i];
            cc[4*i+0] = t.x; cc[4*i+1] = t.y; cc[4*i+2] = t.z; cc[4*i+3] = t.w;
        }
        const float4* dq4 = (const float4*)(depths + rayG * SAMPLES + sb);
        float4 dA = dq4[0], dB = dq4[1];
        float dd[8] = {dA.x, dA.y, dA.z, dA.w, dB.x, dB.y, dB.z, dB.w};

        float w[8];
        #pragma unroll
        for (int j = 0; j < 8; ++j) {
            const int a   = rLocal * LDS_STRIDE + sb + j;
            const float q  = ldsQ [a];
            const float lv = ldsLv[a];
            const float alpha = 1.0f - q;
            // exclusive transmittance = exp(inclusive_cumsum - own log term)
            const float wt = alpha * __expf(c[j] - lv);
            w[j] = wt;
            rgb0 += wt * cc[3*j+0];
            rgb1 += wt * cc[3*j+1];
            rgb2 += wt * cc[3*j+2];
            dAcc += wt * dd[j];
        }
        float4* wout = (float4*)(outWeights + rayG * SAMPLES + sb);
        wout[0] = make_float4(w[0], w[1], w[2], w[3]);
        wout[1] = make_float4(w[4], w[5], w[6], w[7]);

        T = blockTot;
    }

    const float ft = __expf(T);

    // combine the lane pair (each holds half the samples of the same ray)
    rgb0 += __shfl_xor(rgb0, 16, 32);
    rgb1 += __shfl_xor(rgb1, 16, 32);
    rgb2 += __shfl_xor(rgb2, 16, 32);
    dAcc += __shfl_xor(dAcc, 16, 32);

    if (!hi) {
        outRgb[rayG * 3 + 0] = rgb0;
        outRgb[rayG * 3 + 1] = rgb1;
        outRgb[rayG * 3 + 2] = rgb2;
        outDepth[rayG] = dAcc;
        outFt[rayG]    = ft;
    }
}

extern "C" void kernel_launch(void* const* d_in, const int* in_sizes, int n_in,
                              void* d_out, int out_size, void* d_ws, size_t ws_size,
                              hipStream_t stream) {
    const float* colors    = (const float*)d_in[0];
    const float* densities = (const float*)d_in[1];
    const float* depths    = (const float*)d_in[2];

    const long long nray = (long long)in_sizes[1] / SAMPLES;  // densities: [B,R,S,1]

    float* out        = (float*)d_out;
    float* outRgb     = out;                         // [nray,3]
    float* outDepth   = out + nray * 3;              // [nray]
    float* outWeights = out + nray * 4;              // [nray,64]
    float* outFt      = out + nray * 4 + nray * 64;  // [nray]

    const int blocks = (int)(nray / RAYS_PER_BLOCK); // 131072/64 = 2048, exact
    raymarch_wmma_kernel<<<blocks, 128, 0, stream>>>(
        colors, densities, depths, outRgb, outDepth, outWeights, outFt);
}